// IterNormRotation_36223754174794
// MI455X (gfx1250) — compile-verified
//
#include <hip/hip_runtime.h>
#include <math.h>
#include <stdint.h>

// ---------------------------------------------------------------------------
// IterNorm + rotation for MI455X (gfx1250), f32 WMMA path with CDNA5
// async global->LDS double buffering on the two big GEMMs.
//   out = A @ x - A@mean,  A = sqrt(rTr) * R @ P_10,  P via Newton-Schulz on
//   Sigma_N = (x x^T/m - mean mean^T + eps I) * rTr
// ---------------------------------------------------------------------------

typedef __attribute__((ext_vector_type(2))) float v2f;
typedef __attribute__((ext_vector_type(8))) float v8f;

#define CCH 256
#define HWSZ 3136          // 56*56
#define NB 64
#define MTOT (NB * HWSZ)   // 200704
#define EPSV 1e-5f
#define LDP 68             // padded LDS row stride (floats), 16B-aligned, conflict-free

__device__ __forceinline__ v8f wmma_f32(v2f a, v2f b, v8f c) {
  // V_WMMA_F32_16X16X4_F32 : D = A(16x4) * B(4x16) + C(16x16), full f32
  return __builtin_amdgcn_wmma_f32_16x16x4_f32(false, a, false, b, (short)0, c,
                                               false, false);
}

// CDNA5 async copy: memory -> LDS directly (no VGPR staging), ASYNCcnt-tracked.
// GVS mode: mem_addr = SGPR64(base) + VGPR_I32(voff_bytes); VDST = LDS byte addr.
__device__ __forceinline__ void async_ld_b128(uint32_t lds_addr, const void* base,
                                              uint32_t voff_bytes) {
  asm volatile("global_load_async_to_lds_b128 %0, %1, %2"
               :
               : "v"(lds_addr), "v"(voff_bytes), "s"(base)
               : "memory");
}
__device__ __forceinline__ void wait_async_le8() {
  asm volatile("s_wait_asynccnt 8" ::: "memory");
}
__device__ __forceinline__ void wait_async_0() {
  asm volatile("s_wait_asynccnt 0" ::: "memory");
}
__device__ __forceinline__ uint32_t lds_off(const void* p) {
  return (uint32_t)(uintptr_t)p;   // low 32 bits of generic LDS ptr = LDS byte addr
}

// ---------------------------------------------------------------------------
__global__ void zero_kernel(float* p, int n) {
  int i = blockIdx.x * blockDim.x + threadIdx.x;
  if (i < n) p[i] = 0.0f;
}

// one block (256 threads) per channel: mean[c] = sum_{b,hw} X[b][c][hw] / m
__global__ void mean_kernel(const float* __restrict__ X, float* __restrict__ mean) {
  const int c = blockIdx.x;
  const int t = threadIdx.x;
  float s = 0.0f;
  for (int b = 0; b < NB; ++b) {
    const float* row = X + ((size_t)b * CCH + c) * HWSZ;
    for (int i = t; i < HWSZ; i += 256) s += row[i];
  }
  __shared__ float red[256];
  red[t] = s;
  __syncthreads();
  for (int off = 128; off > 0; off >>= 1) {
    if (t < off) red[t] += red[t + off];
    __syncthreads();
  }
  if (t == 0) mean[c] = red[0] * (1.0f / (float)MTOT);
}

// ---------------------------------------------------------------------------
// Gram: G[i][j] += sum_hw X[b][i][hw] * X[b][j][hw]  (K split over batches)
// block = 256 threads (8 waves), one 64x64 C-tile per batch image.
// Async double-buffered global->LDS: fetch of chunk k+1 overlaps WMMA on k.
__global__ void gram_kernel(const float* __restrict__ X, float* __restrict__ G) {
  const int tile = blockIdx.x;            // 0..15 -> 4x4 tiles of 64
  const int ci0 = (tile >> 2) * 64;
  const int cj0 = (tile & 3) * 64;
  const int bb = blockIdx.y;
  const int t = threadIdx.x;
  const int lane = t & 31, wave = t >> 5;
  const int l = lane & 15, half = lane >> 4;
  const int trow = wave & 3;              // 16-row sub-tile
  const int cbase = (wave >> 2) * 32;     // 0 or 32: two 16-col sub-tiles

  __shared__ float Ai[2][64 * LDP];       // [ci][k]
  __shared__ float Aj[2][64 * LDP];       // [cj][k]

  v8f acc0 = {}; v8f acc1 = {};
  const int r  = t >> 2;                  // 0..63 load row
  const int cg = (t & 3) * 16;            // 4 x b128 per thread per tile

  const uint32_t vbaseA = (uint32_t)((((ci0 + r) * HWSZ) + cg) * 4);
  const uint32_t vbaseB = (uint32_t)((((cj0 + r) * HWSZ) + cg) * 4);
  const float* base = X + (size_t)bb * CCH * HWSZ;   // uniform -> SGPR pair

  // issue chunk kc into buffer s: 8 async b128 ops per wave
#define GRAM_ISSUE(kc, s)                                                   \
  {                                                                         \
    uint32_t la = lds_off(&Ai[s][r * LDP + cg]);                            \
    uint32_t lb = lds_off(&Aj[s][r * LDP + cg]);                            \
    uint32_t va = vbaseA + (uint32_t)((kc) * 4);                            \
    uint32_t vb = vbaseB + (uint32_t)((kc) * 4);                            \
    _Pragma("unroll") for (int i = 0; i < 4; ++i) {                         \
      async_ld_b128(la + 16u * i, base, va + 16u * i);                      \
      async_ld_b128(lb + 16u * i, base, vb + 16u * i);                      \
    }                                                                       \
  }

  GRAM_ISSUE(0, 0)
  int s = 0;
  for (int kc = 0; kc < HWSZ; kc += 64, s ^= 1) {
    if (kc + 64 < HWSZ) {
      GRAM_ISSUE(kc + 64, s ^ 1)
      wait_async_le8();                   // previous chunk's 8 ops complete
    } else {
      wait_async_0();
    }
    __syncthreads();                      // all waves' chunk data in LDS
    const float* Aw = Ai[s];
    const float* Bw = Aj[s];
#pragma unroll
    for (int kk = 0; kk < 64; kk += 4) {
      v2f a, b0, b1;
      a.x  = Aw[(trow * 16 + l) * LDP + kk + 2 * half];
      a.y  = Aw[(trow * 16 + l) * LDP + kk + 1 + 2 * half];
      b0.x = Bw[(cbase + l) * LDP + kk + 2 * half];
      b0.y = Bw[(cbase + l) * LDP + kk + 1 + 2 * half];
      b1.x = Bw[(cbase + 16 + l) * LDP + kk + 2 * half];
      b1.y = Bw[(cbase + 16 + l) * LDP + kk + 1 + 2 * half];
      acc0 = wmma_f32(a, b0, acc0);
      acc1 = wmma_f32(a, b1, acc1);
    }
    __syncthreads();                      // done with buffer s before reuse
  }
#undef GRAM_ISSUE
#pragma unroll
  for (int rr = 0; rr < 8; ++rr) {
    const int gi = ci0 + trow * 16 + rr + 8 * half;
    atomicAdd(&G[gi * CCH + cj0 + cbase + l], acc0[rr]);
    atomicAdd(&G[gi * CCH + cj0 + cbase + 16 + l], acc1[rr]);
  }
}

// ---------------------------------------------------------------------------
// single block: Sigma_N = (G/m - mean mean^T + eps I) * rTr ; P = I ; scal[0]=sqrt(rTr)
__global__ void finalize_sigma_kernel(const float* __restrict__ G,
                                      const float* __restrict__ mean,
                                      float* __restrict__ SigN,
                                      float* __restrict__ P,
                                      float* __restrict__ scal) {
  const int t = threadIdx.x;
  const float invm = 1.0f / (float)MTOT;
  const float mt = mean[t];
  for (int i = 0; i < 256; ++i) {
    const int idx = i * 256 + t;
    float v = G[idx] * invm - mean[i] * mt;
    if (i == t) v += EPSV;
    SigN[idx] = v;
  }
  __syncthreads();
  __shared__ float red[256];
  __shared__ float rtr_s;
  red[t] = SigN[t * 256 + t];
  __syncthreads();
  for (int off = 128; off > 0; off >>= 1) {
    if (t < off) red[t] += red[t + off];
    __syncthreads();
  }
  if (t == 0) {
    rtr_s = 1.0f / red[0];
    scal[0] = sqrtf(rtr_s);
  }
  __syncthreads();
  const float rtr = rtr_s;
  for (int i = 0; i < 256; ++i) {
    const int idx = i * 256 + t;
    SigN[idx] *= rtr;
    P[idx] = (i == t) ? 1.0f : 0.0f;
  }
}

// ---------------------------------------------------------------------------
// generic 256x256x256 matmul: D = alpha*(alpha_ptr?*alpha_ptr:1)*(A@B) + beta*Cin
// grid = 32 blocks x 256 threads; one 16x16 WMMA tile per wave (L2-hot operands)
__global__ void mm256_kernel(const float* __restrict__ A, const float* __restrict__ B,
                             const float* __restrict__ Cin, float* __restrict__ D,
                             float alpha, float beta, const float* alpha_ptr) {
  const int t = threadIdx.x;
  const int lane = t & 31, wavel = t >> 5;
  const int w = blockIdx.x * 8 + wavel;   // 0..255 tile id
  const int row0 = (w >> 4) * 16;
  const int col0 = (w & 15) * 16;
  const int l = lane & 15, half = lane >> 4;
  v8f acc = {};
  for (int k = 0; k < 256; k += 4) {
    v2f a, b;
    a.x = A[(row0 + l) * 256 + k + 2 * half];
    a.y = A[(row0 + l) * 256 + k + 1 + 2 * half];
    b.x = B[(k + 2 * half) * 256 + col0 + l];
    b.y = B[(k + 1 + 2 * half) * 256 + col0 + l];
    acc = wmma_f32(a, b, acc);
  }
  const float ae = alpha * (alpha_ptr ? alpha_ptr[0] : 1.0f);
#pragma unroll
  for (int rr = 0; rr < 8; ++rr) {
    const int gi = (row0 + rr + 8 * half) * 256 + col0 + l;
    float v = ae * acc[rr];
    if (Cin) v += beta * Cin[gi];
    D[gi] = v;
  }
}

// bias[d] = sum_c Amat[d][c] * mean[c]
__global__ void bias_kernel(const float* __restrict__ Amat,
                            const float* __restrict__ mean,
                            float* __restrict__ bias) {
  const int d = threadIdx.x;
  float s = 0.0f;
  for (int c = 0; c < 256; ++c) s += Amat[d * 256 + c] * mean[c];
  bias[d] = s;
}

// ---------------------------------------------------------------------------
// out[b][d][hw] = sum_c Amat[d][c] * X[b][c][hw] - bias[d]
// block = 256 threads, 64d x 64hw tile, K=256 via async double-buffered LDS.
__global__ void apply_kernel(const float* __restrict__ X,
                             const float* __restrict__ Amat,
                             const float* __restrict__ bias,
                             float* __restrict__ Out) {
  const int hw0 = blockIdx.x * 64;        // 49 tiles of 64 == 3136
  const int d0 = blockIdx.y * 64;         // 4 tiles
  const int bb = blockIdx.z;              // 64 images
  const int t = threadIdx.x;
  const int lane = t & 31, wave = t >> 5;
  const int l = lane & 15, half = lane >> 4;
  const int trow = wave & 3;
  const int cbase = (wave >> 2) * 32;

  __shared__ float lA[2][64 * LDP];       // [d][k]
  __shared__ float lX[2][64 * LDP];       // [c][hw]

  v8f acc0 = {}, acc1 = {};
  const int r  = t >> 2;
  const int cg = (t & 3) * 16;

  const uint32_t vbaseA = (uint32_t)(((d0 + r) * 256 + cg) * 4);
  const float* baseX = X + ((size_t)bb * CCH) * HWSZ;   // uniform -> SGPR pair
  const uint32_t vbaseX = (uint32_t)((r * HWSZ + hw0 + cg) * 4);

#define APPLY_ISSUE(kc, s)                                                  \
  {                                                                         \
    uint32_t la = lds_off(&lA[s][r * LDP + cg]);                            \
    uint32_t lx = lds_off(&lX[s][r * LDP + cg]);                            \
    uint32_t va = vbaseA + (uint32_t)((kc) * 4);                            \
    uint32_t vx = vbaseX + (uint32_t)((kc) * HWSZ * 4);                     \
    _Pragma("unroll") for (int i = 0; i < 4; ++i) {                         \
      async_ld_b128(la + 16u * i, Amat, va + 16u * i);                      \
      async_ld_b128(lx + 16u * i, baseX, vx + 16u * i);                     \
    }                                                                       \
  }

  APPLY_ISSUE(0, 0)
  int s = 0;
  for (int kc = 0; kc < 256; kc += 64, s ^= 1) {
    if (kc + 64 < 256) {
      APPLY_ISSUE(kc + 64, s ^ 1)
      wait_async_le8();
    } else {
      wait_async_0();
    }
    __syncthreads();
    const float* Aw = lA[s];
    const float* Xw = lX[s];
#pragma unroll
    for (int kk = 0; kk < 64; kk += 4) {
      v2f a, b0, b1;
      a.x  = Aw[(trow * 16 + l) * LDP + kk + 2 * half];
      a.y  = Aw[(trow * 16 + l) * LDP + kk + 1 + 2 * half];
      b0.x = Xw[(kk + 2 * half) * LDP + cbase + l];
      b0.y = Xw[(kk + 1 + 2 * half) * LDP + cbase + l];
      b1.x = Xw[(kk + 2 * half) * LDP + cbase + 16 + l];
      b1.y = Xw[(kk + 1 + 2 * half) * LDP + cbase + 16 + l];
      acc0 = wmma_f32(a, b0, acc0);
      acc1 = wmma_f32(a, b1, acc1);
    }
    __syncthreads();
  }
#undef APPLY_ISSUE
#pragma unroll
  for (int rr = 0; rr < 8; ++rr) {
    const int d = d0 + trow * 16 + rr + 8 * half;
    const float bv = bias[d];
    const size_t basep = ((size_t)bb * CCH + d) * HWSZ + hw0;
    Out[basep + cbase + l] = acc0[rr] - bv;
    Out[basep + cbase + 16 + l] = acc1[rr] - bv;
  }
}

// ---------------------------------------------------------------------------
extern "C" void kernel_launch(void* const* d_in, const int* in_sizes, int n_in,
                              void* d_out, int out_size, void* d_ws, size_t ws_size,
                              hipStream_t stream) {
  (void)in_sizes; (void)n_in; (void)out_size; (void)ws_size;
  const float* X = (const float*)d_in[0];        // [64,256,56,56]
  const float* R = (const float*)d_in[1];        // [1,256,256]
  float* Out = (float*)d_out;
  float* ws = (float*)d_ws;

  float* mean = ws;               // 256
  float* bias = ws + 256;         // 256
  float* scal = ws + 512;         // few scalars
  float* G    = ws + 1024;        // 65536
  float* SigN = G    + 65536;
  float* P0   = SigN + 65536;
  float* T1   = P0   + 65536;
  float* T2   = T1   + 65536;
  float* T3   = T2   + 65536;
  float* Amat = T3   + 65536;

  // 1. zero Gram accumulator (atomic-accumulated)
  zero_kernel<<<dim3(256), dim3(256), 0, stream>>>(G, 65536);
  // 2. per-channel mean
  mean_kernel<<<dim3(256), dim3(256), 0, stream>>>(X, mean);
  // 3. Gram matrix G = x x^T (WMMA + async LDS double buffering)
  gram_kernel<<<dim3(16, 64), dim3(256), 0, stream>>>(X, G);
  // 4. Sigma_N, rTr, P=I
  finalize_sigma_kernel<<<dim3(1), dim3(256), 0, stream>>>(G, mean, SigN, P0, scal);

  // 5. Newton-Schulz: P <- 1.5 P - 0.5 P^3 Sigma_N   (10 iterations)
  float* P = P0;
  float* S = T1;
  for (int it = 0; it < 10; ++it) {
    mm256_kernel<<<dim3(32), dim3(256), 0, stream>>>(P,  P,    nullptr, T2, 1.0f,  0.0f, nullptr);
    mm256_kernel<<<dim3(32), dim3(256), 0, stream>>>(T2, P,    nullptr, T3, 1.0f,  0.0f, nullptr);
    mm256_kernel<<<dim3(32), dim3(256), 0, stream>>>(T3, SigN, P,       S, -0.5f,  1.5f, nullptr);
    float* tmp = P; P = S; S = tmp;
  }

  // 6. Amat = sqrt(rTr) * R @ P  (device-side scale via scal[0])
  mm256_kernel<<<dim3(32), dim3(256), 0, stream>>>(R, P, nullptr, Amat, 1.0f, 0.0f, scal);
  // 7. bias = Amat @ mean
  bias_kernel<<<dim3(1), dim3(256), 0, stream>>>(Amat, mean, bias);
  // 8. out = Amat @ x - bias  (big WMMA GEMM, async LDS double buffering)
  apply_kernel<<<dim3(49, 4, 64), dim3(256), 0, stream>>>(X, Amat, bias, Out);
}